// TopologyDiceLossOriginal_5574867550370
// MI455X (gfx1250) — compile-verified
//
#include <hip/hip_runtime.h>
#include <hip/hip_bf16.h>
#include <stdint.h>

// Problem constants
#define S   128
#define S2  (S * S)
#define S3  (S * S * S)
#define B   8
#define ROWF 132                 // padded floats per LDS row (528 B: 16B-aligned, bank-friendly)
#define TILEF (S * ROWF)         // 16896 floats per staged slice
#define NEG_BIG (-1.0e30f)

// ---- CDNA5 async global->LDS helpers (gfx1250) --------------------------
// LDS byte offset of a generic pointer into shared memory
__device__ __forceinline__ uint32_t lds_off(const void* p) {
  return (uint32_t)(uintptr_t)(__attribute__((address_space(3))) const void*)p;
}

// 16 bytes per lane, global -> LDS, tracked by ASYNCcnt
__device__ __forceinline__ void async_copy_b128(uint32_t lds_byte_off, const void* gsrc) {
  uint64_t ga = (uint64_t)(uintptr_t)gsrc;
  asm volatile("global_load_async_to_lds_b128 %0, %1, off"
               :: "v"(lds_byte_off), "v"(ga)
               : "memory");
}

__device__ __forceinline__ void wait_async_zero() {
  asm volatile("s_wait_asynccnt 0" ::: "memory");
}

// ---- Kernel 0: zero the double accumulators (24 dice + 5 topo) ----------
__global__ void kzero(double* p) {
  if (threadIdx.x < 29) p[threadIdx.x] = 0.0;
}

// ---- Kernel A: (b,y) slices -> P0 (max over z), P1 (max over x), dice ---
__global__ __launch_bounds__(256) void kernelA(const float* __restrict__ in,
                                               const float* __restrict__ tg,
                                               float* __restrict__ P,
                                               float* __restrict__ G,
                                               double* __restrict__ acc) {
  extern __shared__ float smem[];
  float* shA = smem;              // yh slice: [x][z] padded rows
  float* shT = smem + TILEF;      // yt slice
  float* red = smem + 2 * TILEF;  // 24 floats of reduce scratch

  const int blk  = blockIdx.x;        // b*S + y
  const int b    = blk >> 7;
  const int y    = blk & (S - 1);
  const int tid  = threadIdx.x;
  const int lane = tid & 31;
  const int wave = tid >> 5;          // 0..7 (wave32)

  const size_t sliceBase = (size_t)(b * 2 + 1) * S3 + (size_t)y * S;
  const float* gA = in + sliceBase;   // rows at + x*S2, contiguous in z
  const float* gT = tg + sliceBase;

  const uint32_t ldsA0 = lds_off(shA);
  const uint32_t ldsT0 = lds_off(shT);

  // Each wave async-loads 16 rows of each tensor: 32 lanes x 16B = one 512B row.
  for (int r = wave; r < S; r += 8) {
    const float* srcA = gA + (size_t)r * S2 + lane * 4;
    const float* srcT = gT + (size_t)r * S2 + lane * 4;
    uint32_t dA = ldsA0 + (uint32_t)(r * ROWF * 4 + lane * 16);
    uint32_t dT = ldsT0 + (uint32_t)(r * ROWF * 4 + lane * 16);
    async_copy_b128(dA, srcA);
    async_copy_b128(dT, srcT);
  }
  wait_async_zero();
  __syncthreads();

  // P0[b][x][y] = max over z (row scan). Threads 0..127 -> yh, 128..255 -> yt.
  {
    int x = tid & (S - 1);
    const float* row = (tid < S) ? (shA + x * ROWF) : (shT + x * ROWF);
    float m = NEG_BIG;
    #pragma unroll 4
    for (int z = 0; z < S; ++z) m = fmaxf(m, row[z]);
    float* dst = (tid < S) ? P : G;
    dst[((0 * B + b) * S + x) * S + y] = m;
  }

  // P1[b][y][z] = max over x (column scan; ROWF padding keeps it conflict-free).
  {
    int z = tid & (S - 1);
    const float* base = (tid < S) ? shA : shT;
    float m = NEG_BIG;
    #pragma unroll 4
    for (int x = 0; x < S; ++x) m = fmaxf(m, base[x * ROWF + z]);
    float* dst = (tid < S) ? P : G;
    dst[((1 * B + b) * S + y) * S + z] = m;
  }

  // Dice partials over the slice: sum(a*t), sum(a*a), sum(t*t)
  float ia = 0.f, aa = 0.f, tt = 0.f;
  for (int e = tid; e < S2; e += 256) {
    int x = e >> 7, z = e & (S - 1);
    float a = shA[x * ROWF + z];
    float t = shT[x * ROWF + z];
    ia = fmaf(a, t, ia);
    aa = fmaf(a, a, aa);
    tt = fmaf(t, t, tt);
  }
  for (int o = 16; o > 0; o >>= 1) {
    ia += __shfl_down(ia, o);
    aa += __shfl_down(aa, o);
    tt += __shfl_down(tt, o);
  }
  if (lane == 0) { red[wave] = ia; red[8 + wave] = aa; red[16 + wave] = tt; }
  __syncthreads();
  if (tid == 0) {
    float sia = 0.f, saa = 0.f, stt = 0.f;
    for (int w = 0; w < 8; ++w) { sia += red[w]; saa += red[8 + w]; stt += red[16 + w]; }
    atomicAdd(&acc[b],      (double)sia);
    atomicAdd(&acc[8 + b],  (double)saa);
    atomicAdd(&acc[16 + b], (double)stt);
  }
}

// ---- Kernel B: (b,x) slices -> P2 (max over y); coalesced column reads --
__global__ __launch_bounds__(128) void kernelB(const float* __restrict__ in,
                                               const float* __restrict__ tg,
                                               float* __restrict__ P,
                                               float* __restrict__ G) {
  const int blk = blockIdx.x;         // b*S + x
  const int b   = blk >> 7;
  const int x   = blk & (S - 1);
  const int z   = threadIdx.x;        // 0..127, contiguous across lanes

  const size_t base = (size_t)(b * 2 + 1) * S3 + (size_t)x * S2 + z;
  const float* gA = in + base;
  const float* gT = tg + base;

  float ma = NEG_BIG, mt = NEG_BIG;
  #pragma unroll 4
  for (int y = 0; y < S; ++y) {
    ma = fmaxf(ma, gA[(size_t)y * S]);
    mt = fmaxf(mt, gT[(size_t)y * S]);
  }
  P[((2 * B + b) * S + x) * S + z] = ma;
  G[((2 * B + b) * S + x) * S + z] = mt;
}

// ---- Kernel C: multi-scale max-pool + |Gk - Pk| accumulation ------------
__global__ __launch_bounds__(256) void kernelC(const float* __restrict__ P,
                                               const float* __restrict__ G,
                                               double* __restrict__ loss) {
  __shared__ float red[8];
  const int kidx = blockIdx.x >> 3;    // 0..4
  const int b    = blockIdx.x & 7;
  const int k    = 2 << kidx;          // 2,4,8,16,32
  const int n    = S / k;
  const int cells = n * n;

  float local = 0.f;
  for (int c = threadIdx.x; c < cells; c += 256) {
    int i = c / n, j = c % n;
    float pk = 0.f, gk = 0.f;
    for (int t = 0; t < 3; ++t) {
      const float* Pp = P + (size_t)(((t * B + b) * S + i * k) * S + j * k);
      const float* Gp = G + (size_t)(((t * B + b) * S + i * k) * S + j * k);
      float mp = NEG_BIG, mg = NEG_BIG;
      for (int di = 0; di < k; ++di)
        for (int dj = 0; dj < k; ++dj) {
          mp = fmaxf(mp, Pp[di * S + dj]);
          mg = fmaxf(mg, Gp[di * S + dj]);
        }
      pk += mp; gk += mg;
    }
    local += fabsf(gk - pk);
  }
  for (int o = 16; o > 0; o >>= 1) local += __shfl_down(local, o);
  const int lane = threadIdx.x & 31, wave = threadIdx.x >> 5;
  if (lane == 0) red[wave] = local;
  __syncthreads();
  if (threadIdx.x == 0) {
    float s = 0.f;
    for (int w = 0; w < 8; ++w) s += red[w];
    atomicAdd(&loss[kidx], (double)s);
  }
}

// ---- Kernel D: finalize scalar ------------------------------------------
__global__ void kernelD(const double* __restrict__ acc,
                        const double* __restrict__ loss,
                        float* __restrict__ out) {
  if (threadIdx.x == 0 && blockIdx.x == 0) {
    double topo = 0.0;
    for (int kidx = 0; kidx < 5; ++kidx) {
      int k = 2 << kidx;
      int n = S / k;
      topo += loss[kidx] / (double)(B * n * n);
    }
    topo *= (1.0 / 5.0);
    double dm = 0.0;
    for (int bb = 0; bb < B; ++bb) {
      double inter = acc[bb], as = acc[8 + bb], ts = acc[16 + bb];
      dm += (2.0 * inter + 1e-6) / (as + ts);
    }
    dm *= (1.0 / (double)B);
    out[0] = (float)(topo + (1.0 - dm));
  }
}

extern "C" void kernel_launch(void* const* d_in, const int* in_sizes, int n_in,
                              void* d_out, int out_size, void* d_ws, size_t ws_size,
                              hipStream_t stream) {
  (void)in_sizes; (void)n_in; (void)out_size; (void)ws_size;
  const float* in = (const float*)d_in[0];
  const float* tg = (const float*)d_in[1];

  // ws layout: P (3*8*128*128 f32) | G (same) | 29 doubles (dice 24 + topo 5)
  float*  P    = (float*)d_ws;
  float*  G    = P + 3 * B * S2;
  double* acc  = (double*)((char*)d_ws + (size_t)2 * 3 * B * S2 * sizeof(float));
  double* loss = acc + 24;

  kzero<<<1, 32, 0, stream>>>(acc);

  const size_t shbytes = (size_t)(2 * TILEF + 32) * sizeof(float); // ~135.3 KB
  kernelA<<<B * S, 256, shbytes, stream>>>(in, tg, P, G, acc);
  kernelB<<<B * S, 128, 0, stream>>>(in, tg, P, G);
  kernelC<<<5 * B, 256, 0, stream>>>(P, G, loss);
  kernelD<<<1, 32, 0, stream>>>(acc, loss, (float*)d_out);
}